// FVMemoryBank_73650099192086
// MI455X (gfx1250) — compile-verified
//
#include <hip/hip_runtime.h>
#include <hip/hip_bf16.h>
#include <cstdint>

// ---------------------------------------------------------------------------
// FVMemoryBank update for MI455X (gfx1250).
//
// Work: out[0:64M)  = view1_mem with rows y <- l2norm(0.5*old + 0.5*l2norm(audio))
//       out[64M:128M) = view2_mem likewise with video.
// 1.03 GB of traffic, ~5 MFLOP -> pure HBM-bandwidth problem (~44us @ 23.3TB/s).
// Bulk copy uses the CDNA5 async data-mover path (global_load_async_to_lds_b128 /
// global_store_async_from_lds_b128, ASYNCcnt) with 8 outstanding 16B transfers
// per lane per phase to cover HBM latency. Row updates use one wave32 per row
// with __shfl_xor butterflies for the sum-of-squares reductions.
// ---------------------------------------------------------------------------

#define ASYNC_WAIT0() asm volatile("s_wait_asynccnt 0" ::: "memory")

constexpr int COPY_U = 8;   // outstanding async B128 transfers per lane per phase

// Copy both memory banks into d_out through the async LDS bounce.
// src0 -> dst[0 : n4each), src1 -> dst[n4each : 2*n4each)   (units: float4)
__global__ __launch_bounds__(256) void bank_copy_async(
    const float4* __restrict__ src0, const float4* __restrict__ src1,
    float4* __restrict__ dst, long n4each)
{
    __shared__ float4 stage[256 * COPY_U];   // 32 KB / block (<< 320 KB WGP LDS)

    const int  tid    = threadIdx.x;
    const long stride = (long)gridDim.x * blockDim.x;
    const long total  = 2 * n4each;

    // Low 32 bits of a generic shared-aperture address are the LDS byte offset
    // (ISA 10.2); each lane owns COPY_U 16B slots.
    const unsigned lds_base = (unsigned)(size_t)(&stage[tid * COPY_U]);

    for (long i = (long)blockIdx.x * blockDim.x + tid; i < total; i += stride * COPY_U) {
        // Phase 1: queue up to COPY_U async 16B loads (ASYNCcnt tracks them).
#pragma unroll
        for (int u = 0; u < COPY_U; ++u) {
            const long idx = i + (long)u * stride;
            if (idx < total) {
                const float4* sp = (idx < n4each) ? (src0 + idx) : (src1 + (idx - n4each));
                const unsigned lds_off = lds_base + (unsigned)u * 16u;
                asm volatile("global_load_async_to_lds_b128 %0, %1, off"
                             :: "v"(lds_off), "v"(sp)
                             : "memory");
            }
        }
        ASYNC_WAIT0();
        // Phase 2: queue the matching async 16B stores LDS -> dst.
#pragma unroll
        for (int u = 0; u < COPY_U; ++u) {
            const long idx = i + (long)u * stride;
            if (idx < total) {
                float4* dp = dst + idx;
                const unsigned lds_off = lds_base + (unsigned)u * 16u;
                asm volatile("global_store_async_from_lds_b128 %0, %1, off"
                             :: "v"(dp), "v"(lds_off)
                             : "memory");
            }
        }
        ASYNC_WAIT0();   // LDS slots must drain before reuse next iteration
    }
}

// One wave32 per (bank,row): lane owns 4 consecutive floats of the 128-dim row.
__global__ __launch_bounds__(256) void scatter_update(
    const float* __restrict__ audio, const float* __restrict__ video,
    const float* __restrict__ mem0,  const float* __restrict__ mem1,
    const int*   __restrict__ y,     float* __restrict__ out,
    int bs, long mem_elems)
{
    const int lane = threadIdx.x & 31;
    const int wid  = blockIdx.x * (blockDim.x >> 5) + (threadIdx.x >> 5);
    if (wid >= 2 * bs) return;

    const int bank = (wid >= bs) ? 1 : 0;      // 0: view1/audio, 1: view2/video
    const int row  = bank ? (wid - bs) : wid;

    const float* emb = bank ? video : audio;
    const float* mem = bank ? mem1  : mem0;
    float*       dst = out + (bank ? mem_elems : 0);

    const int tgt = y[row];

    const float4 e = *(const float4*)(emb + (size_t)row * 128 + lane * 4);
    const float4 m = *(const float4*)(mem + (size_t)tgt * 128 + lane * 4);

    // ||emb_row||  (wave32 butterfly allreduce)
    float s = e.x * e.x + e.y * e.y + e.z * e.z + e.w * e.w;
#pragma unroll
    for (int off = 16; off > 0; off >>= 1) s += __shfl_xor(s, off, 32);
    const float n1 = fmaxf(sqrtf(s), 1e-12f);

    // l = 0.5 * mem_row + 0.5 * (emb_row / ||emb_row||)   (momentum = 0.5 both)
    float4 l;
    l.x = 0.5f * m.x + 0.5f * (e.x / n1);
    l.y = 0.5f * m.y + 0.5f * (e.y / n1);
    l.z = 0.5f * m.z + 0.5f * (e.z / n1);
    l.w = 0.5f * m.w + 0.5f * (e.w / n1);

    // ||l||
    float s2 = l.x * l.x + l.y * l.y + l.z * l.z + l.w * l.w;
#pragma unroll
    for (int off = 16; off > 0; off >>= 1) s2 += __shfl_xor(s2, off, 32);
    const float n2 = fmaxf(sqrtf(s2), 1e-12f);

    float4 r;
    r.x = l.x / n2;  r.y = l.y / n2;  r.z = l.z / n2;  r.w = l.w / n2;
    *(float4*)(dst + (size_t)tgt * 128 + lane * 4) = r;
}

extern "C" void kernel_launch(void* const* d_in, const int* in_sizes, int n_in,
                              void* d_out, int out_size, void* d_ws, size_t ws_size,
                              hipStream_t stream) {
    const float* audio = (const float*)d_in[0];   // (bs, 128) f32
    const float* video = (const float*)d_in[1];   // (bs, 128) f32
    const float* mem0  = (const float*)d_in[2];   // (M, 128) f32
    const float* mem1  = (const float*)d_in[3];   // (M, 128) f32
    const int*   y     = (const int*)d_in[4];     // (bs,) i32

    float* out = (float*)d_out;                   // [new_view1 | new_view2]

    const int  bs        = in_sizes[4];           // 4096
    const long mem_elems = (long)in_sizes[2];     // M*128 = 64,000,000
    const long n4each    = mem_elems / 4;         // float4 count per bank

    // Bulk copy of both banks (1 GB of traffic) -> saturate HBM via async DMA.
    bank_copy_async<<<2048, 256, 0, stream>>>(
        (const float4*)mem0, (const float4*)mem1, (float4*)out, n4each);

    // Tiny scatter (4 MB of traffic): 2*bs wave32's, 8 waves per block.
    const int waves   = 2 * bs;
    const int sblocks = (waves + 7) / 8;
    scatter_update<<<sblocks, 256, 0, stream>>>(
        audio, video, mem0, mem1, y, out, bs, mem_elems);
}